// YoloLayer_88184268521540
// MI455X (gfx1250) — compile-verified
//
#include <hip/hip_runtime.h>

// YOLO decode layer for MI455X (gfx1250).
// Bandwidth-bound elementwise transform + channel->position transpose.
// Strategy: coalesced b128 loads, VALU sigmoid/exp, LDS transpose staging,
// then CDNA5 async stores from LDS (global_store_async_from_lds_b128).

#define CH   21          // NUM_CLASSES + 6
#define NA   18          // anchors
#define NB   16          // batch
#define GRID 64
#define GG   4096        // 64*64
#define WPB  4           // waves per block
#define PPL  4           // positions per lane
#define PPW  128         // positions per wave (32 lanes * 4)
#define TPI  32          // tiles per (b,a) plane: 4096/128
#define WAVE_LDS_F (PPW * CH)      // 2688 floats per wave
#define WAVE_LDS_B (WAVE_LDS_F * 4) // 10752 bytes per wave (multiple of 512)

__device__ __forceinline__ float fexp(float x) {
  return __builtin_amdgcn_exp2f(x * 1.44269504088896340736f);
}
__device__ __forceinline__ float fsig(float x) {
  return __builtin_amdgcn_rcpf(1.0f + fexp(-x));
}

__global__ void __launch_bounds__(WPB * 32)
yolo_decode(const float* __restrict__ in,
            const float* __restrict__ anchors,
            float* __restrict__ out) {
  __shared__ __align__(16) float lds[WPB][WAVE_LDS_F];  // 43008 B

  const uint32_t lane = threadIdx.x & 31u;
  // readfirstlane so everything derived from the wave id is provably uniform
  const uint32_t w = (uint32_t)__builtin_amdgcn_readfirstlane((int)(threadIdx.x >> 5));
  const uint32_t t    = blockIdx.x * WPB + w;     // global wave-tile id (uniform)
  const uint32_t tile = t & (TPI - 1);
  const uint32_t a    = (t / TPI) % NA;
  const uint32_t b    = t / (TPI * NA);

  const uint32_t s0 = tile * PPW;                 // wave's first spatial index
  const uint32_t sl = s0 + lane * PPL;            // lane's first spatial index

  // per-wave scalar anchor params
  const float aw8 = anchors[a * 3 + 0] * 8.0f;
  const float ah8 = anchors[a * 3 + 1] * 8.0f;
  const float aa  = anchors[a * 3 + 2];

  // ---- coalesced b128 loads: 21 channels x 4 spatial positions ----
  const float* ip = in + ((size_t)(b * (NA * CH) + a * CH)) * GG + sl;
  float vv[CH][PPL];
#pragma unroll
  for (int c = 0; c < CH; ++c) {
    float4 q = *(const float4*)(ip + (size_t)c * GG);
    vv[c][0] = q.x; vv[c][1] = q.y; vv[c][2] = q.z; vv[c][3] = q.w;
  }

  // ---- transform + transpose into LDS in output (position-major) order ----
  float* wl = &lds[w][0];
#pragma unroll
  for (int k = 0; k < PPL; ++k) {
    const uint32_t s  = sl + k;
    const float    gx = (float)(s & (GRID - 1));
    const float    gy = (float)(s >> 6);
    float* o = wl + (lane * PPL + k) * CH;
    o[0] = (fsig(vv[0][k]) * 1.05f - 0.025f + gx) * 8.0f;  // bx*stride
    o[1] = (fsig(vv[1][k]) * 1.05f - 0.025f + gy) * 8.0f;  // by*stride
    o[2] = fexp(vv[2][k]) * aw8;                           // bw*stride
    o[3] = fexp(vv[3][k]) * ah8;                           // bh*stride
    o[4] = vv[4][k] + aa;                                  // angle
#pragma unroll
    for (int c = 5; c < CH; ++c)                           // conf + 15 classes
      o[c] = fsig(vv[c][k]);
  }

  // DS writes must land in LDS before the async engine reads them
  // (ASYNC ops are NOT ordered with DScnt-tracked LDS ops).
  asm volatile("s_wait_dscnt 0" ::: "memory");

  // ---- async drain: 21 x 512B contiguous, 16B-aligned b128 stores ----
  // out byte base = (b*73728 + a*4096 + s0) * 84, provably multiple of 16.
  const float*   op      = out + ((size_t)(b * (NA * GG) + a * GG + s0)) * CH;
  const uint32_t ldsbase = (uint32_t)(uintptr_t)wl;  // low 32 bits = LDS offset

#pragma unroll
  for (uint32_t r = 0; r < WAVE_LDS_B / 512; ++r) {
    uint32_t goff = r * 512u + lane * 16u;   // global byte offset (VADDR, GVS)
    uint32_t doff = ldsbase + goff;          // LDS byte address (VSRC)
    asm volatile("global_store_async_from_lds_b128 %0, %1, %2"
                 :
                 : "v"(goff), "v"(doff), "s"(op)
                 : "memory");
  }
  // ensure LDS is not deallocated / reused before async stores complete
  asm volatile("s_wait_asynccnt 0x0" ::: "memory");
}

extern "C" void kernel_launch(void* const* d_in, const int* in_sizes, int n_in,
                              void* d_out, int out_size, void* d_ws, size_t ws_size,
                              hipStream_t stream) {
  const float* in      = (const float*)d_in[0];
  const float* anchors = (const float*)d_in[1];
  float*       out     = (float*)d_out;

  const int total_wave_tiles = NB * NA * TPI;   // 16*18*32 = 9216
  dim3 grid(total_wave_tiles / WPB);            // 2304 blocks
  dim3 block(WPB * 32);                         // 128 threads = 4 waves
  hipLaunchKernelGGL(yolo_decode, grid, block, 0, stream, in, anchors, out);
}